// SPR_GCN_88648124990768
// MI455X (gfx1250) — compile-verified
//
#include <hip/hip_runtime.h>
#include <hip/hip_bf16.h>

// ---------------------------------------------------------------------------
// GCN forward for MI455X (gfx1250, wave32).
//   h = embed[x]; h = relu(gcn(h,W1,b1)); h = relu(gcn(h,W2,b2));
//   g = segment_mean(h, batch); out = g@Wlin + blin
// Dense transforms use V_WMMA_F32_16X16X4_F32 (exact f32 semantics).
// W is staged in LDS pair-transposed so each B fragment is one ds_load_b64.
// ---------------------------------------------------------------------------

#define N_NODES   200000
#define N_EDGES   1600000
#define NUM_GRAPHS 2000
#define D          64
#define NUM_CLS    2

typedef __attribute__((ext_vector_type(2))) float v2f;
typedef __attribute__((ext_vector_type(8))) float v8f;

// ---------------- degree / normalization -----------------------------------

__global__ void k_fill1(float* __restrict__ deg, int n) {
    int i = blockIdx.x * blockDim.x + threadIdx.x;
    if (i < n) deg[i] = 1.0f;            // self-loop contributes 1
}

__global__ void k_deg_edges(const int* __restrict__ dst, float* __restrict__ deg, int e) {
    int i = blockIdx.x * blockDim.x + threadIdx.x;
    if (i < e) atomicAdd(&deg[dst[i]], 1.0f);
}

__global__ void k_rsqrt(float* __restrict__ deg, int n) {
    int i = blockIdx.x * blockDim.x + threadIdx.x;
    if (i < n) deg[i] = rsqrtf(deg[i]);
}

// ---------------- WMMA GEMM: t[N,64] = A[N,64] @ W[64,64] -------------------
// One wave per 16-row tile. A fragment per k-block (K=4):
//   lanes 0-15 : {A[m][k0+0], A[m][k0+1]},  lanes 16-31 : {A[m][k0+2], A[m][k0+3]}
// B fragment (4x16): lanes 0-15 rows k0+0/k0+1, lanes 16-31 rows k0+2/k0+3,
//   N = lane%16. LDS holds W pair-transposed: Wp[k/2][n] = {W[k][n], W[k+1][n]}
//   so a B fragment is a single aligned 8-byte LDS load per n-tile.
// C/D: VGPR j -> M = j (lanes 0-15) / M = 8+j (lanes 16-31), N = lane%16.

#define GEMM_BODY(LOAD_A)                                                       \
    __shared__ v2f Wlds[(D / 2) * D];  /* [32 k-pairs][64 n] */                 \
    for (int i = threadIdx.x; i < (D / 2) * D; i += 256) {                      \
        int kp = i >> 6, n = i & 63;                                            \
        v2f w; w.x = W[(2 * kp) * D + n]; w.y = W[(2 * kp + 1) * D + n];        \
        Wlds[i] = w;                                                            \
    }                                                                           \
    __syncthreads();                                                            \
    int wave = (blockIdx.x * 256 + threadIdx.x) >> 5;                           \
    if (wave >= nTiles) return;                                                 \
    int lane = threadIdx.x & 31;                                                \
    int r    = lane & 15;                                                       \
    int half = lane >> 4;                                                       \
    int m0   = wave * 16;                                                       \
    v8f c0 = {}, c1 = {}, c2 = {}, c3 = {};                                     \
    for (int kb = 0; kb < 16; ++kb) {                                           \
        int k = kb * 4 + half * 2;                                              \
        v2f a; LOAD_A;                                                          \
        const v2f* wb = &Wlds[(kb * 2 + half) * D + r];                         \
        v2f b0 = wb[0], b1 = wb[16], b2 = wb[32], b3 = wb[48];                  \
        c0 = __builtin_amdgcn_wmma_f32_16x16x4_f32(false, a, false, b0, (short)0, c0, false, false); \
        c1 = __builtin_amdgcn_wmma_f32_16x16x4_f32(false, a, false, b1, (short)0, c1, false, false); \
        c2 = __builtin_amdgcn_wmma_f32_16x16x4_f32(false, a, false, b2, (short)0, c2, false, false); \
        c3 = __builtin_amdgcn_wmma_f32_16x16x4_f32(false, a, false, b3, (short)0, c3, false, false); \
    }                                                                           \
    float* orow = out + (m0 + half * 8) * D;                                    \
    for (int j = 0; j < 8; ++j) {                                               \
        orow[j * D + r]      = c0[j];                                           \
        orow[j * D + 16 + r] = c1[j];                                           \
        orow[j * D + 32 + r] = c2[j];                                           \
        orow[j * D + 48 + r] = c3[j];                                           \
    }

// Layer 1: A row = embed_table[x[m]] (embedding gather fused, no h0 buffer).
__global__ __launch_bounds__(256) void k_gemm_embed(
        const int* __restrict__ x, const float* __restrict__ Etab,
        const float* __restrict__ W, float* __restrict__ out, int nTiles) {
    GEMM_BODY({
        const float* arow = Etab + x[m0 + r] * D;
        a.x = arow[k]; a.y = arow[k + 1];
    })
}

// Layer 2: A row = relu(in[m]) (in-place safe: tile rows read before stored).
__global__ __launch_bounds__(256) void k_gemm_relu(
        const float* __restrict__ in, const float* __restrict__ W,
        float* __restrict__ out, int nTiles) {
    GEMM_BODY({
        const float* arow = in + (m0 + r) * D;
        a.x = fmaxf(arow[k], 0.0f); a.y = fmaxf(arow[k + 1], 0.0f);
    })
}

// ---------------- aggregation ----------------------------------------------
// agg[i] = dinv[i]^2 * t[i] + b   (self-loop term + bias, pure write = init)
__global__ void k_agg_init(const float* __restrict__ t, const float* __restrict__ dinv,
                           const float* __restrict__ bias, float* __restrict__ agg, int n) {
    int tid = blockIdx.x * blockDim.x + threadIdx.x;
    int i = tid >> 4;
    if (i >= n) return;
    int f = (tid & 15) * 4;
    float di = dinv[i];
    float w  = di * di;
    float4 v  = *reinterpret_cast<const float4*>(t + i * D + f);
    float4 bb = *reinterpret_cast<const float4*>(bias + f);
    float4 o;
    o.x = w * v.x + bb.x; o.y = w * v.y + bb.y;
    o.z = w * v.z + bb.z; o.w = w * v.w + bb.w;
    *reinterpret_cast<float4*>(agg + i * D + f) = o;
}

// agg[dst] += dinv[src]*dinv[dst] * t[src] ; 16 lanes per edge, float4 each.
__global__ void k_agg_edges(const int* __restrict__ src, const int* __restrict__ dst,
                            const float* __restrict__ dinv, const float* __restrict__ t,
                            float* __restrict__ agg, int e) {
    int tid = blockIdx.x * blockDim.x + threadIdx.x;
    int ei = tid >> 4;
    if (ei >= e) return;
    int f = (tid & 15) * 4;
    int s = src[ei], d = dst[ei];
    float w = dinv[s] * dinv[d];
    float4 v = *reinterpret_cast<const float4*>(t + s * D + f);
    float* p = agg + d * D + f;
    atomicAdd(p + 0, w * v.x);
    atomicAdd(p + 1, w * v.y);
    atomicAdd(p + 2, w * v.z);
    atomicAdd(p + 3, w * v.w);
}

// ---------------- pooling + classifier -------------------------------------

__global__ void k_zero(float* __restrict__ p, int n) {
    int i = blockIdx.x * blockDim.x + threadIdx.x;
    if (i < n) p[i] = 0.0f;
}

__global__ void k_pool(const float* __restrict__ h, const int* __restrict__ batch,
                       float* __restrict__ sums, float* __restrict__ cnts, int n) {
    int tid = blockIdx.x * blockDim.x + threadIdx.x;
    int i = tid >> 4;
    if (i >= n) return;
    int f = (tid & 15) * 4;
    int g = batch[i];
    float4 v = *reinterpret_cast<const float4*>(h + i * D + f);
    float* p = sums + g * D + f;
    atomicAdd(p + 0, fmaxf(v.x, 0.0f));
    atomicAdd(p + 1, fmaxf(v.y, 0.0f));
    atomicAdd(p + 2, fmaxf(v.z, 0.0f));
    atomicAdd(p + 3, fmaxf(v.w, 0.0f));
    if ((tid & 15) == 0) atomicAdd(&cnts[g], 1.0f);
}

__global__ void k_final(const float* __restrict__ sums, const float* __restrict__ cnts,
                        const float* __restrict__ Wlin, const float* __restrict__ blin,
                        float* __restrict__ out, int g) {
    int i = blockIdx.x * blockDim.x + threadIdx.x;
    if (i >= g) return;
    float inv = 1.0f / fmaxf(cnts[i], 1.0f);
    float a0 = blin[0], a1 = blin[1];
    const float* s = sums + i * D;
    for (int d = 0; d < D; ++d) {
        float v = s[d] * inv;
        a0 += v * Wlin[d * 2 + 0];
        a1 += v * Wlin[d * 2 + 1];
    }
    out[i * 2 + 0] = a0;
    out[i * 2 + 1] = a1;
}

// ---------------- driver ----------------------------------------------------

extern "C" void kernel_launch(void* const* d_in, const int* in_sizes, int n_in,
                              void* d_out, int out_size, void* d_ws, size_t ws_size,
                              hipStream_t stream) {
    const int*   x     = (const int*)  d_in[0];
    const int*   eidx  = (const int*)  d_in[1];   // [2, E]: src row then dst row
    /* d_in[2] edge_type: unused by reference */
    const int*   batch = (const int*)  d_in[3];
    const float* Etab  = (const float*)d_in[4];
    const float* W1    = (const float*)d_in[5];
    const float* b1    = (const float*)d_in[6];
    const float* W2    = (const float*)d_in[7];
    const float* b2    = (const float*)d_in[8];
    const float* Wlin  = (const float*)d_in[9];
    const float* blin  = (const float*)d_in[10];
    float* out = (float*)d_out;

    const int N = N_NODES, E = N_EDGES, G = NUM_GRAPHS;
    const int* src = eidx;
    const int* dst = eidx + E;

    // workspace layout (floats)
    float* bufT = (float*)d_ws;             // [N, 64] transformed features
    float* bufA = bufT + (size_t)N * D;     // [N, 64] aggregated features
    float* dinv = bufA + (size_t)N * D;     // [N]
    float* sums = dinv + N;                 // [G, 64]
    float* cnts = sums + (size_t)G * D;     // [G]

    const int B = 256;
    int nTiles = N / 16;                       // 12500 (N divisible by 16)
    int gemmBlocks = (nTiles + 7) / 8;         // 8 waves per 256-thread block
    int nodeElemBlocks = (N * 16 + B - 1) / B; // 16 threads per node (float4)
    int edgeElemBlocks = (E * 16 + B - 1) / B;

    // degree -> dinv
    k_fill1    <<<(N + B - 1) / B, B, 0, stream>>>(dinv, N);
    k_deg_edges<<<(E + B - 1) / B, B, 0, stream>>>(dst, dinv, E);
    k_rsqrt    <<<(N + B - 1) / B, B, 0, stream>>>(dinv, N);

    // layer 1: t = embed[x] @ W1 ; agg = norm-scatter + b1
    k_gemm_embed<<<gemmBlocks, B, 0, stream>>>(x, Etab, W1, bufT, nTiles);
    k_agg_init  <<<nodeElemBlocks, B, 0, stream>>>(bufT, dinv, b1, bufA, N);
    k_agg_edges <<<edgeElemBlocks, B, 0, stream>>>(src, dst, dinv, bufT, bufA, E);

    // layer 2: t = relu(agg) @ W2 (in-place) ; agg2 into bufT
    k_gemm_relu<<<gemmBlocks, B, 0, stream>>>(bufA, W2, bufA, nTiles);
    k_agg_init <<<nodeElemBlocks, B, 0, stream>>>(bufA, dinv, b2, bufT, N);
    k_agg_edges<<<edgeElemBlocks, B, 0, stream>>>(src, dst, dinv, bufA, bufT, E);

    // mean-pool (relu fused) + classifier
    k_zero <<<(G * D + G + B - 1) / B, B, 0, stream>>>(sums, G * D + G);
    k_pool <<<nodeElemBlocks, B, 0, stream>>>(bufT, batch, sums, cnts, N);
    k_final<<<(G + B - 1) / B, B, 0, stream>>>(sums, cnts, Wlin, blin, out, G);
}